// AnchorHead_50689204027618
// MI455X (gfx1250) — compile-verified
//
#include <hip/hip_runtime.h>
#include <hip/hip_bf16.h>
#include <math.h>

typedef __attribute__((ext_vector_type(2)))  float v2f;
typedef __attribute__((ext_vector_type(8)))  float v8f;

#define HH 208
#define WWID 336
#define AA 3
#define CC 256
#define NPIX (HH * WWID)           // 69888
#define NANCH (NPIX * AA)          // 209664 per image
#define BB 2
#define MTOT (BB * NPIX)           // 139776 rows of the GEMM
#define PRE_NMS 2000
#define POST_NMS 1000
#define IOU_THR 0.7f
#define BBOX_CLIP 4.135166556742356f   // log(1000/16)

// ---------------------------------------------------------------------------
// Kernel 1: fused 1x1-conv GEMM via V_WMMA_F32_16X16X4_F32.
// One wave -> 16 rows x 16 cols (cols: 0..2 cls, 3..14 reg, 15 zero pad).
// A frag (16x4 fp32): lane m=lane&15, lane-half selects K-half -> global b64.
// B frag (4x16 fp32): staged in LDS as K-pairs -> conflict-free ds b64.
// ---------------------------------------------------------------------------
__global__ __launch_bounds__(256) void gemm_head_kernel(
    const float* __restrict__ feats, const float* __restrict__ w_cls,
    const float* __restrict__ w_reg, const float* __restrict__ b_cls,
    const float* __restrict__ b_reg, float* __restrict__ cls_out,
    float* __restrict__ bbox_out, float* __restrict__ scores_ws) {
  __shared__ v2f w2[128 * 16];  // [k_pair][n] -> (w[2p][n], w[2p+1][n]), 16KB

  const int tid = threadIdx.x;
  float* wf = (float*)w2;
  for (int idx = tid; idx < 4096; idx += 256) {
    int k = idx >> 4;
    int n = idx & 15;
    float v = (n < 3)  ? w_cls[k * 3 + n]
            : (n < 15) ? w_reg[k * 12 + (n - 3)]
                       : 0.0f;
    wf[(((k >> 1) << 4) + n) * 2 + (k & 1)] = v;
  }
  __syncthreads();

  const int wave = tid >> 5;
  const int lane = tid & 31;
  const int n    = lane & 15;   // output column / A row (M)
  const int hi   = lane >> 4;   // lane-half selects K-half of the quad

  const long m0 = (long)blockIdx.x * 128 + (long)wave * 16;
  const v2f* arow = (const v2f*)(feats + (m0 + n) * CC);

  v8f acc = {0.f, 0.f, 0.f, 0.f, 0.f, 0.f, 0.f, 0.f};
  for (int q = 0; q < 64; ++q) {           // K = 256 = 64 quads of 4
    const int p = 2 * q + hi;              // pair index (k = 2p, 2p+1)
    v2f a = arow[p];                       // A[m][k], A[m][k+1]
    v2f b = w2[p * 16 + n];                // B[k][n], B[k+1][n]
    acc = __builtin_amdgcn_wmma_f32_16x16x4_f32(
        false, a, false, b, (short)0, acc, false, false);
  }

  const float bias = (n < 3)  ? b_cls[n]
                   : (n < 15) ? b_reg[n - 3]
                              : 0.0f;
  for (int r = 0; r < 8; ++r) {
    const long gm = m0 + hi * 8 + r;       // D layout: lane-half selects M-half
    const float v = acc[r] + bias;
    if (n < 3) {
      cls_out[gm * 3 + n]   = v;
      scores_ws[gm * 3 + n] = 1.0f / (1.0f + expf(-v));
    } else if (n < 15) {
      bbox_out[gm * 12 + (n - 3)] = v;
    }
  }
}

// ---------------------------------------------------------------------------
// Kernel 2: box decode + clip (anchors synthesized in-register).
// ---------------------------------------------------------------------------
__global__ __launch_bounds__(256) void decode_kernel(
    const float* __restrict__ bbox_out, const float* __restrict__ img_info,
    float* __restrict__ boxes_ws) {
  const int tid = blockIdx.x * blockDim.x + threadIdx.x;
  if (tid >= BB * NANCH) return;
  const int b  = tid / NANCH;
  const int jj = tid - b * NANCH;
  const int pixel = jj / 3;
  const int a     = jj - pixel * 3;
  const int h = pixel / WWID;
  const int w = pixel - h * WWID;

  const float aspx[3] = {1.0f, 1.4f, 0.7f};
  const float aspy[3] = {1.0f, 0.7f, 1.4f};
  const float acy = (h + 0.5f) * 4.0f;
  const float acx = (w + 0.5f) * 4.0f;
  const float ah = 32.0f * aspy[a];
  const float aw = 32.0f * aspx[a];

  const long gm = (long)b * NPIX + pixel;
  const float* d = bbox_out + gm * 12 + a * 4;
  const float dy = d[0];
  const float dx = d[1];
  const float dh = fminf(fmaxf(d[2], -BBOX_CLIP), BBOX_CLIP);
  const float dw = fminf(fmaxf(d[3], -BBOX_CLIP), BBOX_CLIP);

  const float cy = dy * ah + acy;
  const float cx = dx * aw + acx;
  const float hh = expf(dh) * ah;
  const float ww = expf(dw) * aw;

  const float hmax = img_info[b * 2 + 0];
  const float wmax = img_info[b * 2 + 1];
  float* o = boxes_ws + (size_t)tid * 4;
  o[0] = fminf(fmaxf(cy - 0.5f * hh, 0.0f), hmax);
  o[1] = fminf(fmaxf(cx - 0.5f * ww, 0.0f), wmax);
  o[2] = fminf(fmaxf(cy + 0.5f * hh, 0.0f), hmax);
  o[3] = fminf(fmaxf(cx + 0.5f * ww, 0.0f), wmax);
}

// ---------------------------------------------------------------------------
// Kernel 3: exact top-PRE_NMS per image via 4-pass MSB radix select on
// order-preserving float keys (scores are sigmoid outputs -> positive),
// then LDS-atomic compaction with exact tie handling.
// ---------------------------------------------------------------------------
__global__ __launch_bounds__(256) void topk_kernel(
    const float* __restrict__ scores_ws, const float* __restrict__ boxes_ws,
    float* __restrict__ sel_scores, float* __restrict__ sel_boxes) {
  const int b = blockIdx.x;
  const int tid = threadIdx.x;
  const float* sc = scores_ws + (size_t)b * NANCH;
  const float* bx = boxes_ws + (size_t)b * NANCH * 4;

  __shared__ unsigned hist[256];
  __shared__ unsigned sh_prefix, sh_krem, sh_cnt;
  if (tid == 0) { sh_prefix = 0u; sh_krem = PRE_NMS; }
  __syncthreads();

  for (int pass = 0; pass < 4; ++pass) {
    const int shift = 24 - 8 * pass;
    hist[tid] = 0u;
    __syncthreads();
    const unsigned prefix = sh_prefix;
    for (int i = tid; i < NANCH; i += 256) {
      const unsigned key = __float_as_uint(sc[i]);
      if (pass == 0 || (key >> (shift + 8)) == prefix)
        atomicAdd(&hist[(key >> shift) & 255u], 1u);
    }
    __syncthreads();
    if (tid == 0) {
      const unsigned k = sh_krem;
      unsigned cum = 0;
      int d = 255;
      for (; d > 0; --d) {
        if (cum + hist[d] >= k) break;
        cum += hist[d];
      }
      sh_prefix = (sh_prefix << 8) | (unsigned)d;
      sh_krem = k - cum;
    }
    __syncthreads();
  }

  const unsigned T = sh_prefix;  // threshold key: count(>T) < k <= count(>=T)
  if (tid == 0) sh_cnt = 0u;
  __syncthreads();

  for (int i = tid; i < NANCH; i += 256) {
    if (__float_as_uint(sc[i]) > T) {
      const unsigned pos = atomicAdd(&sh_cnt, 1u);
      sel_scores[b * PRE_NMS + pos] = sc[i];
      float* o = sel_boxes + ((size_t)b * PRE_NMS + pos) * 4;
      o[0] = bx[(size_t)i * 4 + 0];
      o[1] = bx[(size_t)i * 4 + 1];
      o[2] = bx[(size_t)i * 4 + 2];
      o[3] = bx[(size_t)i * 4 + 3];
    }
  }
  __syncthreads();
  for (int i = tid; i < NANCH; i += 256) {
    if (__float_as_uint(sc[i]) == T) {
      const unsigned pos = atomicAdd(&sh_cnt, 1u);
      if (pos < PRE_NMS) {
        sel_scores[b * PRE_NMS + pos] = sc[i];
        float* o = sel_boxes + ((size_t)b * PRE_NMS + pos) * 4;
        o[0] = bx[(size_t)i * 4 + 0];
        o[1] = bx[(size_t)i * 4 + 1];
        o[2] = bx[(size_t)i * 4 + 2];
        o[3] = bx[(size_t)i * 4 + 3];
      }
    }
  }
}

// ---------------------------------------------------------------------------
// Kernel 4: greedy NMS, one workgroup per image, data resident in LDS.
// Fixed POST_NMS iterations, suppressed -> -1, argmax tie-break = lowest
// index (packed u64 key: ordered-float || inverted index).
// ---------------------------------------------------------------------------
__global__ __launch_bounds__(256) void nms_kernel(
    const float* __restrict__ sel_scores, const float* __restrict__ sel_boxes,
    float* __restrict__ rois, float* __restrict__ roi_scores) {
  const int b = blockIdx.x;
  const int tid = threadIdx.x;
  __shared__ float sc[PRE_NMS];
  __shared__ float bxs[PRE_NMS * 4];
  __shared__ unsigned long long red[256];
  __shared__ int sh_j;

  for (int i = tid; i < PRE_NMS; i += 256) sc[i] = sel_scores[b * PRE_NMS + i];
  for (int i = tid; i < PRE_NMS * 4; i += 256)
    bxs[i] = sel_boxes[(size_t)b * PRE_NMS * 4 + i];
  __syncthreads();

  for (int it = 0; it < POST_NMS; ++it) {
    unsigned long long best = 0ull;
    for (int i = tid; i < PRE_NMS; i += 256) {
      const unsigned u = __float_as_uint(sc[i]);
      const unsigned ord = (u & 0x80000000u) ? ~u : (u | 0x80000000u);
      const unsigned long long key =
          ((unsigned long long)ord << 32) | (unsigned)(0x7FFFFFFF - i);
      if (key > best) best = key;
    }
    red[tid] = best;
    __syncthreads();
    for (int s = 128; s > 0; s >>= 1) {
      if (tid < s && red[tid + s] > red[tid]) red[tid] = red[tid + s];
      __syncthreads();
    }
    if (tid == 0) {
      const int j = 0x7FFFFFFF - (int)(unsigned)(red[0] & 0xFFFFFFFFu);
      sh_j = j;
      roi_scores[b * POST_NMS + it] = sc[j];
      float* o = rois + ((size_t)b * POST_NMS + it) * 4;
      o[0] = bxs[j * 4 + 0];
      o[1] = bxs[j * 4 + 1];
      o[2] = bxs[j * 4 + 2];
      o[3] = bxs[j * 4 + 3];
    }
    __syncthreads();
    const int j = sh_j;
    const float jy1 = bxs[j * 4 + 0], jx1 = bxs[j * 4 + 1];
    const float jy2 = bxs[j * 4 + 2], jx2 = bxs[j * 4 + 3];
    const float a1 = (jy2 - jy1) * (jx2 - jx1);
    for (int i = tid; i < PRE_NMS; i += 256) {
      const float y1 = fmaxf(jy1, bxs[i * 4 + 0]);
      const float x1 = fmaxf(jx1, bxs[i * 4 + 1]);
      const float y2 = fminf(jy2, bxs[i * 4 + 2]);
      const float x2 = fminf(jx2, bxs[i * 4 + 3]);
      const float inter = fmaxf(y2 - y1, 0.0f) * fmaxf(x2 - x1, 0.0f);
      const float a2 = (bxs[i * 4 + 2] - bxs[i * 4 + 0]) *
                       (bxs[i * 4 + 3] - bxs[i * 4 + 1]);
      const float iou = inter / (a1 + a2 - inter + 1e-8f);
      if (iou > IOU_THR) sc[i] = -1.0f;
    }
    __syncthreads();
  }
}

// ---------------------------------------------------------------------------
extern "C" void kernel_launch(void* const* d_in, const int* in_sizes, int n_in,
                              void* d_out, int out_size, void* d_ws,
                              size_t ws_size, hipStream_t stream) {
  const float* feats    = (const float*)d_in[0];
  const float* img_info = (const float*)d_in[1];
  const float* w_cls    = (const float*)d_in[2];
  const float* b_cls    = (const float*)d_in[3];
  const float* w_reg    = (const float*)d_in[4];
  const float* b_reg    = (const float*)d_in[5];

  float* out = (float*)d_out;
  float* cls_out    = out;                                   // B*H*W*3
  float* bbox_out   = out + (size_t)BB * NPIX * 3;           // B*H*W*12
  float* rois       = bbox_out + (size_t)BB * NPIX * 12;     // B*1000*4
  float* roi_scores = rois + (size_t)BB * POST_NMS * 4;      // B*1000

  float* ws = (float*)d_ws;
  float* scores_ws  = ws;                                    // B*N
  float* boxes_ws   = scores_ws + (size_t)BB * NANCH;        // B*N*4
  float* sel_scores = boxes_ws + (size_t)BB * NANCH * 4;     // B*2000
  float* sel_boxes  = sel_scores + (size_t)BB * PRE_NMS;     // B*2000*4

  gemm_head_kernel<<<MTOT / 128, 256, 0, stream>>>(
      feats, w_cls, w_reg, b_cls, b_reg, cls_out, bbox_out, scores_ws);

  decode_kernel<<<(BB * NANCH + 255) / 256, 256, 0, stream>>>(
      bbox_out, img_info, boxes_ws);

  topk_kernel<<<BB, 256, 0, stream>>>(scores_ws, boxes_ws, sel_scores,
                                      sel_boxes);

  nms_kernel<<<BB, 256, 0, stream>>>(sel_scores, sel_boxes, rois, roi_scores);
}